// Weight_Filter_90804198572306
// MI455X (gfx1250) — compile-verified
//
#include <hip/hip_runtime.h>

// out[b,n,k] = |x[b,n,k]| + (w[b,k]/(C-1)) * (S[b,n] - C*|x[b,n,k]|)
// S[b,n] = sum_c |x[b,n,c]|.  Pure bandwidth-bound: 512 MB @ 23.3 TB/s ~ 22 us.

typedef __attribute__((ext_vector_type(4))) float v4f;
typedef __attribute__((ext_vector_type(4))) int   v4i;

#define B_DIM   64
#define N_DIM   1024
#define C_DIM   1000
#define NVEC    250                 // C_DIM / 4 float4 per row
#define WAVES   8                   // 256 threads / wave32
#define ROWS_PER_WAVE 4
#define TILE_N  (WAVES * ROWS_PER_WAVE)   // 32 rows per block

__global__ __launch_bounds__(256)
void Weight_Filter_90804198572306_kernel(const float* __restrict__ residual,
                                         const float* __restrict__ weight,
                                         float* __restrict__ out) {
    __shared__ float wlds[1024];    // weight row for this batch (4 KB)

    const int b    = blockIdx.y;
    const int n0   = blockIdx.x * TILE_N;
    const int tid  = threadIdx.x;
    const int lane = tid & 31;
    const int wave = tid >> 5;

    // ---- Broadcast weight[b, :] into LDS (CDNA5 async global->LDS DMA) ----
    const float* wrow = weight + (size_t)b * C_DIM;
#if __has_builtin(__builtin_amdgcn_global_load_async_to_lds_b128)
    if (tid < NVEC) {
        __builtin_amdgcn_global_load_async_to_lds_b128(
            (v4i*)(wrow + tid * 4),          // global source (row of weight[b,:])
            (v4i*)(&wlds[tid * 4]),          // LDS destination
            /*offset=*/0, /*cpol=*/0);
    }
#  if __has_builtin(__builtin_amdgcn_s_wait_asynccnt)
    __builtin_amdgcn_s_wait_asynccnt(0);
#  else
    asm volatile("s_wait_asynccnt 0" ::: "memory");
#  endif
#else
    // Fallback path (also used by the host-side parse): plain vector copy.
    if (tid < NVEC) {
        *(v4f*)(&wlds[tid * 4]) = *(const v4f*)(wrow + tid * 4);
    }
#endif
    __syncthreads();

    const float inv  = 1.0f / (float)(C_DIM - 1);
    const float negC = -(float)C_DIM;
    const v4f* wl4   = (const v4f*)wlds;

    for (int r = 0; r < ROWS_PER_WAVE; ++r) {
        const int n = n0 + wave * ROWS_PER_WAVE + r;
        const size_t rowoff = ((size_t)b * N_DIM + n) * (size_t)C_DIM;
        const v4f* __restrict__ xin = (const v4f*)(residual + rowoff);
        v4f* __restrict__       yot = (v4f*)(out + rowoff);

        // ---- Pass 1: load row once (non-temporal), abs, partial sum ----
        v4f a[8];
        float s = 0.0f;
#pragma unroll
        for (int i = 0; i < 8; ++i) {
            const int idx = i * 32 + lane;
            if (idx < NVEC) {
                v4f x = __builtin_nontemporal_load(&xin[idx]);
                v4f t;
                t.x = __builtin_fabsf(x.x);
                t.y = __builtin_fabsf(x.y);
                t.z = __builtin_fabsf(x.z);
                t.w = __builtin_fabsf(x.w);
                a[i] = t;
                s += (t.x + t.y) + (t.z + t.w);
            } else {
                v4f z = {0.0f, 0.0f, 0.0f, 0.0f};
                a[i] = z;
            }
        }

        // ---- Wave32 all-reduce of the row sum ----
#pragma unroll
        for (int off = 16; off >= 1; off >>= 1)
            s += __shfl_xor(s, off, 32);

        // ---- Pass 2: out = a + (w*inv) * (S - C*a), streamed store ----
#pragma unroll
        for (int i = 0; i < 8; ++i) {
            const int idx = i * 32 + lane;
            if (idx < NVEC) {
                const v4f wv = wl4[idx];
                const v4f t  = a[i];
                v4f o;
                o.x = __builtin_fmaf(wv.x * inv, __builtin_fmaf(negC, t.x, s), t.x);
                o.y = __builtin_fmaf(wv.y * inv, __builtin_fmaf(negC, t.y, s), t.y);
                o.z = __builtin_fmaf(wv.z * inv, __builtin_fmaf(negC, t.z, s), t.z);
                o.w = __builtin_fmaf(wv.w * inv, __builtin_fmaf(negC, t.w, s), t.w);
                __builtin_nontemporal_store(o, &yot[idx]);
            }
        }
    }
}

extern "C" void kernel_launch(void* const* d_in, const int* in_sizes, int n_in,
                              void* d_out, int out_size, void* d_ws, size_t ws_size,
                              hipStream_t stream) {
    const float* residual = (const float*)d_in[0];   // [64, 1024, 1000] f32
    const float* weight   = (const float*)d_in[1];   // [64, 1000] f32
    float* out            = (float*)d_out;           // [64, 1024, 1000] f32

    dim3 grid(N_DIM / TILE_N, B_DIM);                // (32, 64)
    dim3 block(256);
    Weight_Filter_90804198572306_kernel<<<grid, block, 0, stream>>>(residual, weight, out);
}